// Sampler_84722524881118
// MI455X (gfx1250) — compile-verified
//
#include <hip/hip_runtime.h>
#include <stdint.h>

// ---------------------------------------------------------------------------
// Top-p (nucleus) sampling, B=256 rows x V=128000 vocab, gfx1250 (MI455X).
// Sort-free, max-pass-free: threshold search on order-flipped logit bits with
// scale-invariant mass u = exp(x/T).  Three streamed passes of logits + one
// of noise ~= 524 MB -> ~22 us @ 23.3 TB/s (floor ~11 us).
// Pass 1 streams via the Tensor Data Mover (tensor_load_to_lds, TENSORcnt);
// passes 2-3 stream via per-lane async DMA (global_load_async_to_lds_b128,
// ASYNCcnt).  All double buffered through LDS.
// ---------------------------------------------------------------------------

#define BLOCK    512
#define TILE_FLT (BLOCK * 4)   // 2048 floats (8 KB) staged per tile
#define NB       8192          // coarse buckets: flipped bits >> 19 (13 bits)
#define NSUB     8192          // fine buckets: (flipped bits >> 6) & 0x1FFF
#define NWAVES   (BLOCK / 32)

typedef unsigned int u32x4 __attribute__((ext_vector_type(4)));
typedef int          i32x4 __attribute__((ext_vector_type(4)));
typedef int          i32x8 __attribute__((ext_vector_type(8)));

struct __align__(16) Smem {
  float    stage[2][TILE_FLT];  // 16 KB staging (double buffered)
  float    hist[NB];            // 32 KB coarse mass histogram
  float    sub[NSUB];           // 32 KB refinement histogram
  float    gsum[NB / 32];       //  1 KB group partial sums for descending scans
  float    redf[NWAVES];
  int      redi[NWAVES];
  unsigned redu[NWAVES];
  float    target, cumAbove;
  int      b1;
  unsigned kstar, kmax;
};

// Monotone (order-preserving) unsigned key for an IEEE-754 float.
__device__ __forceinline__ unsigned flip_key(float x) {
  unsigned b = __float_as_uint(x);
  return b ^ ((b & 0x80000000u) ? 0xFFFFFFFFu : 0x80000000u);
}

// ---------------- TDM path: one descriptor moves the whole tile ------------
// Degenerate-1D D#: tile == tensor (no OOB), data_size=4B, dims1/2 unused.
__device__ __forceinline__ void tdm_issue(const float* __restrict__ row, int V,
                                          int tile, float* stage_buf) {
  const int rem = V - tile * TILE_FLT;
  const int td0 = rem < TILE_FLT ? rem : TILE_FLT;          // elements (fits 16b)
  const uint64_t ga = (uint64_t)(uintptr_t)(row + (size_t)tile * TILE_FLT);
  const unsigned lds = (unsigned)(uintptr_t)stage_buf;

  u32x4 g0;
  g0[0] = 1u;                                               // count=1, no gather
  g0[1] = lds;                                              // lds_addr (bytes)
  g0[2] = (unsigned)(ga & 0xFFFFFFFFu);                     // global_addr[31:0]
  g0[3] = ((unsigned)(ga >> 32) & 0x01FFFFFFu) | 0x80000000u; // [56:32] | type=2

  i32x8 g1 = {0, 0, 0, 0, 0, 0, 0, 0};
  g1[0] = (int)(2u << 16);                                  // data_size = 4B
  g1[1] = (int)(((unsigned)td0 & 0xFFFFu) << 16);           // tensor_dim0[15:0]
  g1[2] = (int)(((unsigned)td0 >> 16) & 0xFFFFu);           // tensor_dim0[31:16]
  g1[3] = (int)(((unsigned)td0 & 0xFFFFu) << 16);           // tile_dim0
  // tile_dim1/2 = 0 (unused), strides = 0

  i32x4 gz = {0, 0, 0, 0};                                  // groups 2/3 unused
#if defined(__clang_major__) && (__clang_major__ >= 23)
  i32x8 gz8 = {0, 0, 0, 0, 0, 0, 0, 0};
  __builtin_amdgcn_tensor_load_to_lds(g0, g1, gz, gz, gz8, 0);
#else
  __builtin_amdgcn_tensor_load_to_lds(g0, g1, gz, gz, 0);
#endif
}

// ---------------- Async-DMA path: each lane moves 16 bytes -----------------
__device__ __forceinline__ void issue_tile(const float* __restrict__ row, int V,
                                           int tile, float* stage_buf) {
  const int fidx = tile * TILE_FLT + (int)threadIdx.x * 4;
  if (fidx < V) {
    unsigned lds  = (unsigned)(uintptr_t)(stage_buf + (int)threadIdx.x * 4);
    unsigned goff = (unsigned)fidx * 4u;
    uint64_t sa   = (uint64_t)(uintptr_t)row;
    asm volatile("global_load_async_to_lds_b128 %0, %1, %2"
                 :: "v"(lds), "v"(goff), "s"(sa) : "memory");
  }
}

__device__ __forceinline__ void wait_async_all() {
  asm volatile("s_wait_asynccnt 0x0" ::: "memory");
}

// Stream via async-DMA (all waves issue, ASYNCcnt).  Double buffered; the
// per-iteration barrier after the wait both publishes the tile and orders
// buffer reuse (issue of tile t+1 targets the buffer last consumed at t-1,
// whose consumers all passed barrier #t).
template <typename F>
__device__ __forceinline__ void stream_row(const float* __restrict__ row, int V,
                                           float (*stage)[TILE_FLT], F&& consume) {
  const int ntiles = (V + TILE_FLT - 1) / TILE_FLT;
  issue_tile(row, V, 0, stage[0]);
  for (int tl = 0; tl < ntiles; ++tl) {
    wait_async_all();
    __syncthreads();
    if (tl + 1 < ntiles) issue_tile(row, V, tl + 1, stage[(tl + 1) & 1]);
    const float* buf = stage[tl & 1] + (int)threadIdx.x * 4;
    const int base = tl * TILE_FLT + (int)threadIdx.x * 4;
#pragma unroll
    for (int k = 0; k < 4; ++k) {
      const int i = base + k;
      if (i < V) consume(i, buf[k]);
    }
  }
}

// Stream via TDM (wave 0 issues one descriptor per tile, TENSORcnt).
template <typename F>
__device__ __forceinline__ void stream_row_tdm(const float* __restrict__ row, int V,
                                               float (*stage)[TILE_FLT], F&& consume) {
  const int ntiles = (V + TILE_FLT - 1) / TILE_FLT;
  if (threadIdx.x < 32) tdm_issue(row, V, 0, stage[0]);
  for (int tl = 0; tl < ntiles; ++tl) {
    if (threadIdx.x < 32) __builtin_amdgcn_s_wait_tensorcnt(0);
    __syncthreads();
    if (tl + 1 < ntiles && threadIdx.x < 32)
      tdm_issue(row, V, tl + 1, stage[(tl + 1) & 1]);
    const float* buf = stage[tl & 1] + (int)threadIdx.x * 4;
    const int base = tl * TILE_FLT + (int)threadIdx.x * 4;
#pragma unroll
    for (int k = 0; k < 4; ++k) {
      const int i = base + k;
      if (i < V) consume(i, buf[k]);
    }
  }
}

__global__ __launch_bounds__(BLOCK)
void Sampler_84722524881118_kernel(const float* __restrict__ logits,
                                   const float* __restrict__ temps,
                                   const float* __restrict__ topps,
                                   const float* __restrict__ noise,
                                   float* __restrict__ out, int V) {
  __shared__ Smem sm;
  const int b    = blockIdx.x;
  const int t    = threadIdx.x;
  const int lane = t & 31, wv = t >> 5;
  const float* __restrict__ row  = logits + (size_t)b * (size_t)V;
  const float* __restrict__ nrow = noise  + (size_t)b * (size_t)V;
  const float invT = 1.0f / temps[b];

  for (int i = t; i < NB; i += BLOCK) sm.hist[i] = 0.f;
  __syncthreads();

  // ---- Pass 1 (TDM stream): Z = sum(u), coarse mass histogram keyed on
  //      flipped logit bits, and max key (always-keep-top-1 rule) -----------
  float zp = 0.f;
  unsigned kmx = 0u;
  stream_row_tdm(row, V, sm.stage, [&](int i, float v) { (void)i;
    float u = __expf(fminf(v * invT, 87.0f));   // scale-invariant mass
    unsigned f = flip_key(v);
    zp += u;
    kmx = (f > kmx) ? f : kmx;
    atomicAdd(&sm.hist[f >> 19], u);            // ds_add_f32
  });
  for (int o = 16; o; o >>= 1) {
    zp += __shfl_down(zp, o, 32);
    unsigned ok = __shfl_down(kmx, o, 32);
    kmx = (ok > kmx) ? ok : kmx;
  }
  if (lane == 0) { sm.redf[wv] = zp; sm.redu[wv] = kmx; }
  __syncthreads();
  if (t == 0) {
    float Z = 0.f;
    unsigned km = 0u;
    for (int w = 0; w < NWAVES; ++w) {
      Z += sm.redf[w];
      km = (sm.redu[w] > km) ? sm.redu[w] : km;
    }
    sm.target = topps[b] * Z;
    sm.kmax = km;
  }
  __syncthreads();
  const float target = sm.target;

  // ---- Descending scan of coarse histogram: find crossing bucket b1 -------
  if (t < NB / 32) {
    float s = 0.f;
    for (int j = 0; j < 32; ++j) s += sm.hist[t * 32 + j];
    sm.gsum[t] = s;
  }
  __syncthreads();
  if (t == 0) {
    float cum = 0.f;
    int g1 = -1;
    for (int g = NB / 32 - 1; g >= 0; --g) {
      float h = sm.gsum[g];
      if (cum + h > target) { g1 = g; break; }
      cum += h;
    }
    int b1 = 0; float cumAbove = cum;
    if (g1 >= 0) {
      bool hit = false;
      for (int j = 31; j >= 0; --j) {
        float h = sm.hist[g1 * 32 + j];
        if (cum + h > target) { b1 = g1 * 32 + j; cumAbove = cum; hit = true; break; }
        cum += h;
      }
      if (!hit) { b1 = g1 * 32; cumAbove = cum; }
    }
    sm.b1 = b1;
    sm.cumAbove = cumAbove;
  }
  __syncthreads();
  const int b1 = sm.b1;
  for (int i = t; i < NSUB; i += BLOCK) sm.sub[i] = 0.f;
  __syncthreads();

  // ---- Pass 2 (async stream): refine bucket b1 on the next 13 key bits ----
  stream_row(row, V, sm.stage, [&](int i, float v) { (void)i;
    unsigned f = flip_key(v);
    if ((int)(f >> 19) == b1) {
      float u = __expf(fminf(v * invT, 87.0f));
      atomicAdd(&sm.sub[(f >> 6) & (unsigned)(NSUB - 1)], u);
    }
  });
  __syncthreads();
  if (t < NSUB / 32) {
    float s = 0.f;
    for (int j = 0; j < 32; ++j) s += sm.sub[t * 32 + j];
    sm.gsum[t] = s;
  }
  __syncthreads();
  if (t == 0) {
    float cum = sm.cumAbove;
    int g1s = -1;
    for (int g = NSUB / 32 - 1; g >= 0; --g) {
      float h = sm.gsum[g];
      if (cum + h > target) { g1s = g; break; }
      cum += h;
    }
    unsigned kstar;
    if (g1s >= 0) {
      int b2 = g1s * 32;
      for (int j = 31; j >= 0; --j) {
        float h = sm.sub[g1s * 32 + j];
        if (cum + h > target) { b2 = g1s * 32 + j; break; }
        cum += h;
      }
      kstar = ((unsigned)b1 << 13) | (unsigned)b2;   // 26-bit key = f >> 6
    } else {
      kstar = (b1 > 0) ? (((unsigned)b1 << 13) - 1u) : 0u;  // keep all of b1
    }
    const unsigned ktop = sm.kmax >> 6;  // key of the top-1 token
    if (ktop > 0u && kstar >= ktop) kstar = ktop - 1u;  // always keep top-1
    sm.kstar = kstar;
  }
  __syncthreads();
  const unsigned kstar = sm.kstar;

  // ---- Pass 3 (async stream): argmax(u / noise) over kept tokens ----------
  float best = -1.f;
  int bidx = 0x7FFFFFFF;
  stream_row(row, V, sm.stage, [&](int i, float v) {
    unsigned K = flip_key(v) >> 6;
    if (K > kstar) {
      float u = __expf(fminf(v * invT, 87.0f));
      float r = u / nrow[i];
      if (r > best || (r == best && i < bidx)) { best = r; bidx = i; }
    }
  });
  for (int o = 16; o; o >>= 1) {
    float ov = __shfl_down(best, o, 32);
    int   oi = __shfl_down(bidx, o, 32);
    if (ov > best || (ov == best && oi < bidx)) { best = ov; bidx = oi; }
  }
  if (lane == 0) { sm.redf[wv] = best; sm.redi[wv] = bidx; }
  __syncthreads();
  if (t == 0) {
    float bv = sm.redf[0]; int bi = sm.redi[0];
    for (int w = 1; w < NWAVES; ++w) {
      float ov = sm.redf[w]; int oi = sm.redi[w];
      if (ov > bv || (ov == bv && oi < bi)) { bv = ov; bi = oi; }
    }
    out[b] = (float)bi;
  }
}

extern "C" void kernel_launch(void* const* d_in, const int* in_sizes, int n_in,
                              void* d_out, int out_size, void* d_ws, size_t ws_size,
                              hipStream_t stream) {
  (void)n_in; (void)out_size; (void)d_ws; (void)ws_size;
  const float* logits = (const float*)d_in[0];
  const float* temps  = (const float*)d_in[1];
  const float* topps  = (const float*)d_in[2];
  const float* noise  = (const float*)d_in[3];
  float* out = (float*)d_out;
  const int B = in_sizes[1];       // temperatures: [B]
  const int V = in_sizes[0] / B;   // logits: [B, V]
  Sampler_84722524881118_kernel<<<dim3(B), dim3(BLOCK), 0, stream>>>(
      logits, temps, topps, noise, out, V);
}